// MeanAggregator_40613210751310
// MI455X (gfx1250) — compile-verified
//
#include <hip/hip_runtime.h>

// MeanAggregator for MI455X (gfx1250).
// Memory-bound gather-mean: per output row, 11 scattered 512-B feature rows are
// pulled with the CDNA5 async global->LDS path (ASYNCcnt), double-buffered per
// wave, then reduced from LDS and stored as b128 (non-temporal: output is
// write-once, keep the L2 full of the 51 MB feature table instead).
// WMMA is deliberately not used: the dense one-hot GEMM equivalent is
// ~1.28 PFLOP vs ~310 MB of traffic for the gather formulation; on a
// 23.3 TB/s part the gather path wins by >1000x.

#define AS1 __attribute__((address_space(1)))
#define AS3 __attribute__((address_space(3)))

typedef float f4v __attribute__((ext_vector_type(4)));
typedef int   v4i __attribute__((vector_size(4 * sizeof(int))));

namespace {
constexpr int kDim          = 128;           // feature dim
constexpr int kK            = 10;            // neighbours
constexpr int kRows         = kK + 1;        // neighbours + self
constexpr int kRowBytes     = kDim * 4;      // 512 B per feature row
constexpr int kF4PerRow     = kDim / 4;      // 32 float4 per row
constexpr int kWavesPerBlk  = 4;
constexpr int kThreads      = kWavesPerBlk * 32;
constexpr int kBlocks       = 2048;          // 8192 persistent waves
}

template <int N>
__device__ __forceinline__ void wait_asynccnt() {
#if __has_builtin(__builtin_amdgcn_s_wait_asynccnt)
  __builtin_amdgcn_s_wait_asynccnt(N);
#else
  asm volatile("s_wait_asynccnt %0" ::"i"(N) : "memory");
#endif
}

// Copy one 512-B feature row (row-base gsrc) into LDS at ldst, 16 B per lane.
__device__ __forceinline__ void async_copy_row(const float* __restrict__ gsrc,
                                               AS3 char* ldst, int lane) {
  AS1 v4i* g = (AS1 v4i*)((const AS1 char*)gsrc + lane * 16);
  AS3 v4i* l = (AS3 v4i*)(ldst + lane * 16);
#if __has_builtin(__builtin_amdgcn_global_load_async_to_lds_b128)
  __builtin_amdgcn_global_load_async_to_lds_b128(g, l, 0, 0);
#else
  unsigned loff = (unsigned)(unsigned long long)l;
  asm volatile("global_load_async_to_lds_b128 %0, %1, off" ::"v"(loff), "v"(g)
               : "memory");
#endif
}

__global__ __launch_bounds__(kThreads) void mean_agg_kernel(
    const int* __restrict__ nodes, const int* __restrict__ neighbours,
    const float* __restrict__ features, float* __restrict__ out, int batch,
    int stride_rows) {
  // Per-wave double-buffered staging: 2 buffers x 11 rows x 512 B.
  __shared__ f4v stage[kWavesPerBlk * 2 * kRows * kF4PerRow];  // 45056 B

  const int lane  = (int)(threadIdx.x & 31u);
  const int wave  = __builtin_amdgcn_readfirstlane((int)(threadIdx.x >> 5));
  const int gwave = (int)blockIdx.x * kWavesPerBlk + wave;

  AS3 char* my = (AS3 char*)&stage[wave * 2 * kRows * kF4PerRow];

  // Issue the 11 async row-gathers for `row` into buffer `buf`.
  auto issue = [&](int buf, int row) {
    int idx[kRows];
    idx[0] = nodes[row];  // row is wave-uniform -> SMEM scalar loads
#pragma unroll
    for (int j = 0; j < kK; ++j) idx[j + 1] = neighbours[row * kK + j];
    AS3 char* base = my + buf * (kRows * kRowBytes);
#pragma unroll
    for (int j = 0; j < kRows; ++j)
      async_copy_row(features + (size_t)idx[j] * kDim, base + j * kRowBytes,
                     lane);
  };

  const float inv = 1.0f / (float)kRows;

  int row = gwave;
  if (row >= batch) return;
  issue(0, row);
  int buf = 0;
  for (;;) {
    const int nrow = row + stride_rows;
    if (nrow < batch) {
      issue(buf ^ 1, nrow);      // prefetch next row's 11 copies
      wait_asynccnt<kRows>();    // in-order completion: current buf is ready
    } else {
      wait_asynccnt<0>();
    }

    // Reduce 11 rows from LDS; lane handles float4 at column lane*4.
    const f4v* b =
        (const f4v*)&stage[(wave * 2 + buf) * kRows * kF4PerRow];  // ds reads
    f4v acc = b[lane];
#pragma unroll
    for (int j = 1; j < kRows; ++j) acc += b[j * kF4PerRow + lane];
    acc *= inv;
    // Write-once output: non-temporal b128 store so the feature table stays
    // resident in L2/WGP$.
    __builtin_nontemporal_store(
        acc, (f4v*)(out + (size_t)row * kDim + (size_t)lane * 4));

    if (nrow >= batch) break;
    row = nrow;
    buf ^= 1;
  }
}

extern "C" void kernel_launch(void* const* d_in, const int* in_sizes, int n_in,
                              void* d_out, int out_size, void* d_ws,
                              size_t ws_size, hipStream_t stream) {
  (void)n_in; (void)out_size; (void)d_ws; (void)ws_size;
  const int*   nodes      = (const int*)d_in[0];
  const int*   neighbours = (const int*)d_in[1];
  const float* features   = (const float*)d_in[2];
  float*       out        = (float*)d_out;
  const int    batch      = in_sizes[0];

  const int stride_rows = kBlocks * kWavesPerBlk;
  mean_agg_kernel<<<kBlocks, kThreads, 0, stream>>>(nodes, neighbours, features,
                                                    out, batch, stride_rows);
}